// DLRM_44092134261025
// MI455X (gfx1250) — compile-verified
//
#include <hip/hip_runtime.h>
#include <hip/hip_bf16.h>

// D = A*B + C, fp32 WMMA fragments for 16x16x4
typedef __attribute__((ext_vector_type(2))) float v2f;   // A/B frag: 16x4 f32 = 64 elems / 32 lanes
typedef __attribute__((ext_vector_type(8))) float v8f;   // C/D frag: 16x16 f32 = 256 elems / 32 lanes

#define NROWS 64
#define DDIM  256
#define OUT_PER_B 2016   // 64*63/2

// One wave per batch. 8 waves (256 threads) per block.
// For each batch: Gram = X (64x256) * X^T, lower-triangular 16x16 tiles only
// (10 of 16 tiles), accumulated over K=256 in steps of 4 via v_wmma_f32_16x16x4_f32.
// The same per-lane 2-float fragment serves as both the A operand (rows) and the
// B operand (columns of X^T), since the A 16x4 layout and B 4x16 layout coincide
// for a symmetric operand pair.
__global__ __launch_bounds__(256) void gram_tril_wmma(const float* __restrict__ X,
                                                      float* __restrict__ out,
                                                      int nbatch) {
  const int lane = threadIdx.x & 31;
  const int wave = threadIdx.x >> 5;
  const int batch = blockIdx.x * 8 + wave;
  if (batch >= nbatch) return;   // uniform per-wave; EXEC stays all-ones inside

  const int m  = lane & 15;      // row (A) / col (B) within 16-tile
  const int kh = lane >> 4;      // 0 or 1 -> K offset 0 or 2 within the 4-wide K step

  const float* Xb = X + (size_t)batch * (NROWS * DDIM);

  // Base pointers for the 4 row-tiles; each lane reads 2 consecutive floats (b64).
  const float* r0 = Xb + (size_t)( 0 + m) * DDIM + 2 * kh;
  const float* r1 = Xb + (size_t)(16 + m) * DDIM + 2 * kh;
  const float* r2 = Xb + (size_t)(32 + m) * DDIM + 2 * kh;
  const float* r3 = Xb + (size_t)(48 + m) * DDIM + 2 * kh;

  // 10 lower-tri tiles in order (i,j): (0,0)(1,0)(1,1)(2,0)(2,1)(2,2)(3,0)(3,1)(3,2)(3,3)
  v8f acc[10];
#pragma unroll
  for (int t = 0; t < 10; ++t) {
#pragma unroll
    for (int v = 0; v < 8; ++v) acc[t][v] = 0.0f;
  }

#pragma unroll 4
  for (int k = 0; k < DDIM; k += 4) {
    v2f f[4];
    f[0] = *(const v2f*)(r0 + k);
    f[1] = *(const v2f*)(r1 + k);
    f[2] = *(const v2f*)(r2 + k);
    f[3] = *(const v2f*)(r3 + k);

    int t = 0;
#pragma unroll
    for (int i = 0; i < 4; ++i) {
#pragma unroll
      for (int j = 0; j <= i; ++j) {
        // 8 args: (neg_a, A, neg_b, B, c_mod, C, reuse_a, reuse_b)
        acc[t] = __builtin_amdgcn_wmma_f32_16x16x4_f32(
            false, f[i], false, f[j], (short)0, acc[t], false, false);
        ++t;
      }
    }
  }

  // Scatter to packed strict-lower-triangle: out[b][i*(i-1)/2 + j], j < i.
  // C/D layout: VGPR v, lanes 0-15 -> row v, lanes 16-31 -> row v+8; col = lane%16.
  float* outb = out + (size_t)batch * OUT_PER_B;
  int t = 0;
#pragma unroll
  for (int it = 0; it < 4; ++it) {
#pragma unroll
    for (int jt = 0; jt <= it; ++jt, ++t) {
#pragma unroll
      for (int v = 0; v < 8; ++v) {
        const int i = it * 16 + v + 8 * kh;
        const int j = jt * 16 + m;
        if (it != jt) {
          // whole tile is strictly below the diagonal
          outb[(i * (i - 1)) / 2 + j] = acc[t][v];
        } else if (i > j) {
          outb[(i * (i - 1)) / 2 + j] = acc[t][v];
        }
      }
    }
  }
}

extern "C" void kernel_launch(void* const* d_in, const int* in_sizes, int n_in,
                              void* d_out, int out_size, void* d_ws, size_t ws_size,
                              hipStream_t stream) {
  const float* X = (const float*)d_in[0];
  float* out = (float*)d_out;
  const int nbatch = in_sizes[0] / (NROWS * DDIM);   // 8192
  const int blocks = (nbatch + 7) / 8;               // 8 batches (waves) per block
  gram_tril_wmma<<<blocks, 256, 0, stream>>>(X, out, nbatch);
}